// FrechetLoss_8624294331089
// MI455X (gfx1250) — compile-verified
//
#include <hip/hip_runtime.h>
#include <math.h>

typedef __attribute__((ext_vector_type(2))) float v2f;
typedef __attribute__((ext_vector_type(8))) float v8f;

#define NPTS 4096
#define KDIM 128

// ---------------------------------------------------------------------------
// Kernel 1: squared row norms, one wave32 per row (coalesced 128-float rows)
// ---------------------------------------------------------------------------
__global__ void row_sqnorm_kernel(const float* __restrict__ A,
                                  float* __restrict__ nrm, int rows) {
  const int gtid = blockIdx.x * blockDim.x + threadIdx.x;
  const int wave = gtid >> 5;
  const int lane = threadIdx.x & 31;
  if (wave >= rows) return;
  const float* row = A + (size_t)wave * KDIM;
  float s = 0.0f;
#pragma unroll
  for (int k = 0; k < KDIM; k += 32) {
    const float v = row[k + lane];
    s = fmaf(v, v, s);
  }
#pragma unroll
  for (int off = 16; off > 0; off >>= 1) s += __shfl_xor(s, off, 32);
  if (lane == 0) nrm[wave] = s;
}

// ---------------------------------------------------------------------------
// Kernel 2: C[m][n] = sqrt(max(pn[m] + tn[n] - 2 * (pred_m . tgt_n), 0))
// One wave per 16x64 strip: 4 accumulators, A register reused across 4 WMMAs
// per k-step (loads/WMMA: 2 -> 1.25).  V_WMMA_F32_16X16X4_F32.
// A(16x4 f32): lane%16 -> M, VGPR v -> K = v + 2*(lane>>4)
// B(4x16 f32): lane%16 -> N, VGPR v -> K = v + 2*(lane>>4)   (B[k][n]=tgt[n][k])
// C/D(16x16):  lane%16 -> N, VGPR v -> M = v + 8*(lane>>4)
// ---------------------------------------------------------------------------
#define NTPW 4   // 16-wide N tiles per wave

__global__ void __launch_bounds__(256)
cdist_wmma_kernel(const float* __restrict__ pred, const float* __restrict__ tgt,
                  const float* __restrict__ pn, const float* __restrict__ tn,
                  float* __restrict__ C) {
  const int gtid = blockIdx.x * blockDim.x + threadIdx.x;
  const int wave = gtid >> 5;
  const int lane = threadIdx.x & 31;
  const int nstrips = NPTS / (16 * NTPW);   // 64 strips per row of tiles
  const int mt = wave / nstrips;
  const int nt = wave - mt * nstrips;
  const int m0 = mt * 16, n0 = nt * (16 * NTPW);
  const int l16 = lane & 15;
  const int half = lane >> 4;

  const float* pA = pred + (size_t)(m0 + l16) * KDIM + 2 * half;
  const float* pB = tgt  + (size_t)(n0 + l16) * KDIM + 2 * half;

  v8f acc[NTPW];
#pragma unroll
  for (int u = 0; u < NTPW; ++u) acc[u] = (v8f){};

#pragma unroll 4
  for (int kb = 0; kb < KDIM; kb += 4) {
    const v2f a = *(const v2f*)(pA + kb);
#pragma unroll
    for (int u = 0; u < NTPW; ++u) {
      const v2f b = *(const v2f*)(pB + (size_t)(16 * u) * KDIM + kb);
      acc[u] = __builtin_amdgcn_wmma_f32_16x16x4_f32(
          /*neg_a=*/false, a, /*neg_b=*/false, b,
          /*c_mod=*/(short)0, acc[u], /*reuse_a=*/false, /*reuse_b=*/false);
    }
  }

  const int mrow = m0 + 8 * half;
  float pnv[8];
#pragma unroll
  for (int v = 0; v < 8; ++v) pnv[v] = pn[mrow + v];

#pragma unroll
  for (int u = 0; u < NTPW; ++u) {
    const float tnj = tn[n0 + 16 * u + l16];
    float* crow = C + (size_t)mrow * NPTS + n0 + 16 * u + l16;
#pragma unroll
    for (int v = 0; v < 8; ++v) {
      const float d2 = pnv[v] + tnj - 2.0f * acc[u][v];
      crow[(size_t)v * NPTS] = sqrtf(fmaxf(d2, 0.0f));
    }
  }
}

// ---------------------------------------------------------------------------
// Kernel 3: Frechet DP. Single 1024-thread workgroup, 4096 sequential rows.
// Per row: clamp-monoid scan; composition of clip(x,lo,hi) functions:
//   (later o earlier).lo = clip(earlier.lo, later.lo, later.hi)  (same for hi)
// 4 elems/thread -> serial compose -> wave32 shuffle scan -> redundant
// cross-wave scan in every wave (2 barriers/row instead of 3).
// ---------------------------------------------------------------------------
#define DPT 1024
#define EPT (NPTS / DPT)   // 4

__device__ __forceinline__ float clipf(float x, float lo, float hi) {
  return fminf(fmaxf(x, lo), hi);
}

__global__ void __launch_bounds__(DPT)
frechet_dp_kernel(const float* __restrict__ C, float* __restrict__ out) {
  __shared__ float Dbuf[2][NPTS];          // double-buffered DP rows (32 KB)
  __shared__ float wLo[32], wHi[32];       // per-wave clamp composites
  const int t = threadIdx.x;
  const int lane = t & 31;
  const int wid = t >> 5;
  const float INF = __builtin_inff();

  int cur = 0;
  for (int i = 0; i < NPTS; ++i) {
    const float* prevD = Dbuf[cur ^ 1];
    float* curD = Dbuf[cur];
    const int j0 = t * EPT;

    const float4 c4 = *(const float4*)(C + (size_t)i * NPTS + j0);
    if (i + 1 < NPTS)  // hide next row's L2 read under this row's scan
      __builtin_prefetch(C + (size_t)(i + 1) * NPTS + j0, 0, 1);

    float lo[EPT], hi[EPT];
    lo[0] = c4.x; lo[1] = c4.y; lo[2] = c4.z; lo[3] = c4.w;
    if (i == 0) {
      // row 0: D[0,j] = prefix-max(c[0,0..j]) == clamps (c_j,+inf), j=0 pinned
      hi[0] = hi[1] = hi[2] = hi[3] = INF;
      if (t == 0) hi[0] = lo[0];
    } else {
      float pm = (t == 0) ? 0.0f : prevD[j0 - 1];
#pragma unroll
      for (int e = 0; e < EPT; ++e) {
        const float pj = prevD[j0 + e];
        hi[e] = fmaxf(fminf(pj, pm), lo[e]);
        pm = pj;
      }
      if (t == 0) {  // column 0: D[i,0] = max(D[i-1,0], c[i,0]) (constant clamp)
        const float d0 = fmaxf(prevD[0], lo[0]);
        lo[0] = d0; hi[0] = d0;
      }
    }

    // compose this thread's EPT clamps (in j order)
    float Lo = lo[0], Hi = hi[0];
#pragma unroll
    for (int e = 1; e < EPT; ++e) {
      Lo = clipf(Lo, lo[e], hi[e]);
      Hi = clipf(Hi, lo[e], hi[e]);
    }

    // inclusive wave32 scan of clamp composition
    float sLo = Lo, sHi = Hi;
#pragma unroll
    for (int off = 1; off < 32; off <<= 1) {
      const float pLo = __shfl_up(sLo, off, 32);
      const float pHi = __shfl_up(sHi, off, 32);
      if (lane >= off) {
        const float nLo = clipf(pLo, sLo, sHi);
        const float nHi = clipf(pHi, sLo, sHi);
        sLo = nLo; sHi = nHi;
      }
    }
    if (lane == 31) { wLo[wid] = sLo; wHi[wid] = sHi; }
    __syncthreads();

    // every wave redundantly scans the 32 wave composites; each wave then
    // broadcasts its exclusive wave-prefix from lane wid-1 (identity for wave 0)
    float aLo = wLo[lane], aHi = wHi[lane];
#pragma unroll
    for (int off = 1; off < 32; off <<= 1) {
      const float pLo = __shfl_up(aLo, off, 32);
      const float pHi = __shfl_up(aHi, off, 32);
      if (lane >= off) {
        const float nLo = clipf(pLo, aLo, aHi);
        const float nHi = clipf(pHi, aLo, aHi);
        aLo = nLo; aHi = nHi;
      }
    }
    const float wEx = (float)(wid > 0);   // selector (avoid divergent shfl idx<0)
    const int src = (wid > 0) ? (wid - 1) : 0;
    float eLo = __shfl(aLo, src, 32);
    float eHi = __shfl(aHi, src, 32);
    eLo = wEx ? eLo : -INF;               // wave 0: identity clamp
    eHi = wEx ? eHi :  INF;

    // compose with lanes [0,lane) inside this wave
    const float lLo = __shfl_up(sLo, 1, 32);
    const float lHi = __shfl_up(sHi, 1, 32);
    if (lane > 0) {
      eLo = clipf(eLo, lLo, lHi);
      eHi = clipf(eHi, lLo, lHi);
    }

    // evaluate the recurrence across this thread's elements
    float x = clipf(0.0f, eLo, eHi);   // seed irrelevant: col-0 clamp is constant
#pragma unroll
    for (int e = 0; e < EPT; ++e) {
      x = clipf(x, lo[e], hi[e]);
      curD[j0 + e] = x;
    }
    if (i == NPTS - 1 && t == DPT - 1) out[0] = x;  // LOSS_WEIGHT == 1.0
    __syncthreads();
    cur ^= 1;
  }
}

// ---------------------------------------------------------------------------
extern "C" void kernel_launch(void* const* d_in, const int* in_sizes, int n_in,
                              void* d_out, int out_size, void* d_ws, size_t ws_size,
                              hipStream_t stream) {
  const float* pred = (const float*)d_in[0];
  const float* tgt  = (const float*)d_in[1];
  float* ws = (float*)d_ws;
  float* pn = ws;                    // 4096 floats
  float* tn = ws + NPTS;             // 4096 floats
  float* C  = ws + 2 * NPTS;         // 4096*4096 floats (64 MB, L2-resident)

  {  // row norms: one wave per row
    dim3 blk(256);
    dim3 grd((NPTS * 32) / 256);
    row_sqnorm_kernel<<<grd, blk, 0, stream>>>(pred, pn, NPTS);
    row_sqnorm_kernel<<<grd, blk, 0, stream>>>(tgt, tn, NPTS);
  }
  {  // distance matrix via FP32 WMMA: 256 x 64 strips of 16x64, 8 waves/block
    const int nstrips = NPTS / (16 * NTPW);          // 64
    const int waves = (NPTS / 16) * nstrips;         // 16384
    dim3 blk(256);
    dim3 grd(waves / 8);
    cdist_wmma_kernel<<<grd, blk, 0, stream>>>(pred, tgt, pn, tn, C);
  }
  // sequential DP: single workgroup, row-parallel clamp scan
  frechet_dp_kernel<<<1, DPT, 0, stream>>>(C, (float*)d_out);
}